// Spiking1DALIFLayer_8246337208355
// MI455X (gfx1250) — compile-verified
//
#include <hip/hip_runtime.h>
#include <cstdint>
#include <cstddef>

#ifndef __has_builtin
#define __has_builtin(x) 0
#endif

#if defined(__gfx1250__) && __has_builtin(__builtin_amdgcn_global_load_async_to_lds_b128)
#define USE_ASYNC_LDS 1
#else
#define USE_ASYNC_LDS 0
#endif

#define S_ALPHA 0.75f
#define S_BETA  0.65f
#define S_GAMMA 0.75f
#define S_RHO   0.75f

#if defined(__gfx1250__)
typedef int v4i_t __attribute__((__vector_size__(16)));
typedef v4i_t __attribute__((address_space(1)))* gptr_b128;
typedef v4i_t __attribute__((address_space(3)))* lptr_b128;
#endif

template <int N>
__device__ __forceinline__ void wait_asynccnt_le() {
#if defined(__gfx1250__)
#if __has_builtin(__builtin_amdgcn_s_wait_asynccnt)
  __builtin_amdgcn_s_wait_asynccnt(N);
  asm volatile("" ::: "memory");
#else
  asm volatile("s_wait_asynccnt %0" :: "n"(N) : "memory");
#endif
#endif
}

// One ALIF step; returns the new spike and updates state in place.
// Matches: rst = spk*Vth; inp = a*inp + x; mem = (mem-rst)*b + inp;
//          tha = r*tha + spk_old; ath = Vth + g*tha; spk = (mem-ath) > 0
__device__ __forceinline__ float alif_step(float xt, float vth_c,
                                           float& inp, float& mem,
                                           float& spk, float& tha) {
  const float rst = spk * vth_c;
  inp = fmaf(S_ALPHA, inp, xt);
  mem = fmaf(S_BETA, mem - rst, inp);
  tha = fmaf(S_RHO, tha, spk);
  spk = (mem > fmaf(S_GAMMA, tha, vth_c)) ? 1.0f : 0.0f;
  return spk;
}

// Layout: x, out are (B, C, T) with T contiguous; row = b*C + c; T = 1024.
// Block = 128 threads = 4 waves; each wave owns 32 consecutive rows and a
// private TRIPLE-buffered LDS tile (32 rows x 32 timesteps per buffer),
// filled by async global->LDS b128 copies issued 2 chunks ahead (up to 24
// outstanding per wave -> ~12KB in flight per wave; parallelism is capped at
// 2048 waves by the serial time scan, so per-wave MLP must cover the HBM
// latency-bandwidth product). Coalescing: 8 consecutive lanes cover one full
// 128B cacheline of one row. Tile is wave-private -> no barriers, only
// s_wait_asynccnt (async loads complete in order).
__global__ __launch_bounds__(128) void alif_scan_kernel(
    const float* __restrict__ x, const float* __restrict__ vth,
    float* __restrict__ out, int cmask) {
  constexpr int T      = 1024;
  constexpr int CHUNK  = 32;         // timesteps staged per chunk (128B/row)
  constexpr int NCHUNK = T / CHUNK;  // 32
  constexpr int STR    = 36;         // LDS floats per row (144B: 16B aligned, padded)
  constexpr int WAVES  = 4;
  constexpr int NBUF   = 3;

  const int tid      = threadIdx.x;
  const int wave     = tid >> 5;
  const int lane     = tid & 31;
  const int row_base = blockIdx.x * (WAVES * 32) + wave * 32;
  const int row      = row_base + lane;
  const float vth_c  = vth[row & cmask];

  float inp = 0.0f, mem = 0.0f, spk = 0.0f, tha = 0.0f;
  float* __restrict__ orow = out + (size_t)row * T;

#if USE_ASYNC_LDS
  __shared__ float smem[NBUF * WAVES * 32 * STR];  // 54 KB

  auto issue = [&](int c, int buf) {
    const int base = (buf * WAVES + wave) * 32 * STR;
    const float* gbase = x + (size_t)row_base * T + (size_t)c * CHUNK;
#pragma unroll
    for (int l = 0; l < 8; ++l) {
      const int idx = l * 32 + lane;   // 16B unit index within 32x128B tile
      const int r   = idx >> 3;        // row within wave's 32 rows
      const int q   = idx & 7;         // 16B slot within that row's 128B
      const float* g = gbase + (size_t)r * T + q * 4;
      float* s = &smem[base + r * STR + q * 4];
      __builtin_amdgcn_global_load_async_to_lds_b128(
          (gptr_b128)(g), (lptr_b128)(s), 0, 0);
    }
  };

  issue(0, 0);               // prologue: 2 chunks in flight
  issue(1, 1);

  int buf = 0;               // rotating buffer index, = c % NBUF
  int nxt = 2;               // = (c + 2) % NBUF
  for (int c = 0; c < NCHUNK; ++c) {
    if (c + 2 < NCHUNK) {
      issue(c + 2, nxt);     // keep 2 chunks ahead
      wait_asynccnt_le<16>();// chunk c's 8 loads done (in-order completion)
    } else if (c + 1 < NCHUNK) {
      wait_asynccnt_le<8>();
    } else {
      wait_asynccnt_le<0>();
    }
    const float* lrow = &smem[(buf * WAVES + wave) * 32 * STR + lane * STR];
#pragma unroll
    for (int j = 0; j < 8; ++j) {
      const float4 xv = *(const float4*)(lrow + j * 4);
      float4 sv;
      sv.x = alif_step(xv.x, vth_c, inp, mem, spk, tha);
      sv.y = alif_step(xv.y, vth_c, inp, mem, spk, tha);
      sv.z = alif_step(xv.z, vth_c, inp, mem, spk, tha);
      sv.w = alif_step(xv.w, vth_c, inp, mem, spk, tha);
      *(float4*)(orow + c * CHUNK + j * 4) = sv;
    }
    buf = (buf == NBUF - 1) ? 0 : buf + 1;
    nxt = (nxt == NBUF - 1) ? 0 : nxt + 1;
  }
#else
  // Fallback: direct vectorized streaming loads (cache-line reuse across the
  // 8 consecutive b128 loads a thread makes within each 128B of its row).
  const float* __restrict__ xrow = x + (size_t)row * T;
  for (int c = 0; c < NCHUNK; ++c) {
#pragma unroll
    for (int j = 0; j < 8; ++j) {
      const float4 xv = *(const float4*)(xrow + c * CHUNK + j * 4);
      float4 sv;
      sv.x = alif_step(xv.x, vth_c, inp, mem, spk, tha);
      sv.y = alif_step(xv.y, vth_c, inp, mem, spk, tha);
      sv.z = alif_step(xv.z, vth_c, inp, mem, spk, tha);
      sv.w = alif_step(xv.w, vth_c, inp, mem, spk, tha);
      *(float4*)(orow + c * CHUNK + j * 4) = sv;
    }
  }
#endif
}

extern "C" void kernel_launch(void* const* d_in, const int* in_sizes, int n_in,
                              void* d_out, int out_size, void* d_ws, size_t ws_size,
                              hipStream_t stream) {
  const float* x   = (const float*)d_in[0];  // (B, C, T) f32
  const float* vth = (const float*)d_in[1];  // (C,) f32
  float* out       = (float*)d_out;          // (B, C, T) f32

  const int T    = 1024;
  const int C    = in_sizes[1];              // 512 (power of two)
  const int rows = in_sizes[0] / T;          // B*C = 65536

  const int ROWS_PER_BLOCK = 128;            // 4 waves x 32 rows
  dim3 grid(rows / ROWS_PER_BLOCK);
  dim3 block(ROWS_PER_BLOCK);
  alif_scan_kernel<<<grid, block, 0, stream>>>(x, vth, out, C - 1);
}